// WindowSTM_54528904790917
// MI455X (gfx1250) — compile-verified
//
#include <hip/hip_runtime.h>
#include <hip/hip_bf16.h>
#include <math.h>

typedef __attribute__((ext_vector_type(2)))  float v2f;
typedef __attribute__((ext_vector_type(8)))  float v8f;

#define BN_EPS 1e-5f

// async global->LDS builtin takes addrspace-qualified int* (clang prints these
// as '__device__ int*' / '__shared__ int*' in HIP mode)
#define AS1I(p) ((__attribute__((address_space(1))) int*)(p))
#define AS3I(p) ((__attribute__((address_space(3))) int*)(p))

#if defined(__has_builtin)
#if __has_builtin(__builtin_amdgcn_global_load_async_to_lds_b32)
#define HAVE_ASYNC_LDS 1
#endif
#endif

__device__ __forceinline__ void wait_async0() {
#if __has_builtin(__builtin_amdgcn_s_wait_asynccnt)
    __builtin_amdgcn_s_wait_asynccnt(0);
#else
    asm volatile("s_wait_asynccnt 0x0" ::: "memory");
#endif
}

// ---------------------------------------------------------------------------
// fp32 GEMM on the WMMA pipe: C[m,n] = sum_k A[m*lda+k]*B[k*ldb+n].
// 4 waves/block, one 16x16 C tile per wave along n. grid=(N/64, M/16, Z).
// z = blockIdx.z: bi=z/T, ti=z%T; B += bi*sBb+ti*sBt; C += bi*sCb+ti*sCt.
// ---------------------------------------------------------------------------
__global__ __launch_bounds__(128)
void gemm_wmma(const float* __restrict__ A, const float* __restrict__ B,
               float* __restrict__ C, int K, int lda, int ldb, int ldc,
               int T, long sBb, long sBt, long sCb, long sCt)
{
    int z  = blockIdx.z;
    int bi = z / T, ti = z - bi * T;
    B += (size_t)bi * sBb + (size_t)ti * sBt;
    C += (size_t)bi * sCb + (size_t)ti * sCt;

    const int tid   = threadIdx.x;
    const int lane  = tid & 31;
    const int wv    = tid >> 5;
    const int l16   = lane & 15;
    const int hh    = lane >> 4;               // half-wave: 0/1
    const int mBase = blockIdx.y * 16;
    const int nBase = blockIdx.x * 64 + wv * 16;

    v8f acc = {0.f,0.f,0.f,0.f,0.f,0.f,0.f,0.f};

    // A frag: 2 floats/lane, m=l16, k=2*hh+{0,1}; B symmetric (k rows, n cols).
    const float* Arow = A + (size_t)(mBase + l16) * lda + 2 * hh;
    const float* Bcol = B + (size_t)(2 * hh) * ldb + nBase + l16;
    for (int kk = 0; kk < K; kk += 4) {
        v2f a, b;
        a.x = Arow[kk];
        a.y = Arow[kk + 1];
        b.x = Bcol[(size_t)kk * ldb];
        b.y = Bcol[(size_t)(kk + 1) * ldb];
        acc = __builtin_amdgcn_wmma_f32_16x16x4_f32(
                  false, a, false, b, (short)0, acc, false, false);
    }

    // C layout: VGPR r -> row r (lanes 0-15) / r+8 (lanes 16-31), col = lane%16.
    float* Crow = C + (size_t)(mBase + hh * 8) * ldc + nBase + l16;
    #pragma unroll
    for (int r = 0; r < 8; ++r) Crow[(size_t)r * ldc] = acc[r];
}

// ---------------------------------------------------------------------------
// Same GEMM, fused BatchNorm(eval)+ReLU epilogue. z = batch index.
// ---------------------------------------------------------------------------
__global__ __launch_bounds__(128)
void gemm_wmma_bn_relu(const float* __restrict__ A, const float* __restrict__ B,
                       float* __restrict__ C, int K, int lda, int ldb, int ldc,
                       long sBb, long sCb,
                       const float* __restrict__ gamma, const float* __restrict__ beta,
                       const float* __restrict__ mean,  const float* __restrict__ var)
{
    int bi = blockIdx.z;
    B += (size_t)bi * sBb;
    C += (size_t)bi * sCb;

    const int tid   = threadIdx.x;
    const int lane  = tid & 31;
    const int wv    = tid >> 5;
    const int l16   = lane & 15;
    const int hh    = lane >> 4;
    const int mBase = blockIdx.y * 16;
    const int nBase = blockIdx.x * 64 + wv * 16;

    v8f acc = {0.f,0.f,0.f,0.f,0.f,0.f,0.f,0.f};

    const float* Arow = A + (size_t)(mBase + l16) * lda + 2 * hh;
    const float* Bcol = B + (size_t)(2 * hh) * ldb + nBase + l16;
    for (int kk = 0; kk < K; kk += 4) {
        v2f a, b;
        a.x = Arow[kk];
        a.y = Arow[kk + 1];
        b.x = Bcol[(size_t)kk * ldb];
        b.y = Bcol[(size_t)(kk + 1) * ldb];
        acc = __builtin_amdgcn_wmma_f32_16x16x4_f32(
                  false, a, false, b, (short)0, acc, false, false);
    }

    float* Crow = C + (size_t)(mBase + hh * 8) * ldc + nBase + l16;
    #pragma unroll
    for (int r = 0; r < 8; ++r) {
        int   m   = mBase + hh * 8 + r;
        float inv = gamma[m] * rsqrtf(var[m] + BN_EPS);
        float v   = acc[r] * inv + (beta[m] - mean[m] * inv);
        Crow[(size_t)r * ldc] = fmaxf(v, 0.f);
    }
}

// ---------------------------------------------------------------------------
// Fused windowed attention for one (b,h) row. 8 waves.
//   qk: (b,32,1024)  mk: (b,3,32,1024)  mv: (b,3,128,1024)
//   feat: (b,256,1024), writes channels [128,256).
// Per (t,dy) chunk:
//   scores  S[w][u] = Q^T K          (WMMA, waves 0-3, 4 tiles)
//   softmax over w per key column (reference's axis=2), banded weights -> sW
//   D[w][c] += sW * V^T              (WMMA, 8 waves x 2 tiles, persistent C)
// K/V tiles staged via async global->LDS loads (ASYNCcnt) when available.
// ---------------------------------------------------------------------------
__global__ __launch_bounds__(256)
void attn_kernel(const float* __restrict__ qk, const float* __restrict__ mk,
                 const float* __restrict__ mv, float* __restrict__ feat)
{
    const int h    = blockIdx.x;    // 0..31
    const int b    = blockIdx.y;    // 0..1
    const int tid  = threadIdx.x;   // 0..255
    const int lane = tid & 31;
    const int wv   = tid >> 5;      // 0..7
    const int l16  = lane & 15;
    const int hh   = lane >> 4;

    __shared__ float sQ[32][33];    // [c][w]
    __shared__ float sK[32][33];    // [c][u]
    __shared__ float sS[32][33];    // [w][u] scores
    __shared__ float sW[32][33];    // [w][u] banded softmax weights
    __shared__ float sV[128][33];   // [c][u]
    __shared__ float cmax[16];
    __shared__ float crinv[16];

    // Load Q tile once: qk[b, c, h, :]
    {
        const float* qb = qk + (size_t)b * 32 * 1024 + (size_t)h * 32;
        for (int idx = tid; idx < 1024; idx += 256) {
            int c = idx >> 5, ww = idx & 31;
            sQ[c][ww] = qb[(size_t)c * 1024 + ww];
        }
    }

    // Persistent value accumulators: wave wv owns D tiles of D[w(32) x c(128)]:
    // mBase=(wv&1)*16, n tiles (wv>>1)*2+{0,1}.
    v8f vacc0 = {0.f,0.f,0.f,0.f,0.f,0.f,0.f,0.f};
    v8f vacc1 = {0.f,0.f,0.f,0.f,0.f,0.f,0.f,0.f};
    const int mBase = (wv & 1) * 16;
    const int n0    = ((wv >> 1) * 2 + 0) * 16;
    const int n1    = ((wv >> 1) * 2 + 1) * 16;

    __syncthreads();

    for (int tt = 0; tt < 3; ++tt)
    for (int dy = 0; dy < 15; ++dy) {
        const int row = h + dy - 7;
        if (row < 0 || row >= 32) continue;   // padded row: V==0, no contribution

        const float* kb = mk + ((size_t)(b * 3 + tt) * 32)  * 1024 + (size_t)row * 32;
        const float* vb = mv + ((size_t)(b * 3 + tt) * 128) * 1024 + (size_t)row * 32;

#if HAVE_ASYNC_LDS
        for (int idx = tid; idx < 1024; idx += 256) {
            int c = idx >> 5, u = idx & 31;
            __builtin_amdgcn_global_load_async_to_lds_b32(
                AS1I(kb + (size_t)c * 1024 + u), AS3I(&sK[c][u]), 0, 0);
        }
        for (int idx = tid; idx < 4096; idx += 256) {
            int c = idx >> 5, u = idx & 31;
            __builtin_amdgcn_global_load_async_to_lds_b32(
                AS1I(vb + (size_t)c * 1024 + u), AS3I(&sV[c][u]), 0, 0);
        }
        wait_async0();
#else
        for (int idx = tid; idx < 1024; idx += 256) {
            int c = idx >> 5, u = idx & 31;
            sK[c][u] = kb[(size_t)c * 1024 + u];
        }
        for (int idx = tid; idx < 4096; idx += 256) {
            int c = idx >> 5, u = idx & 31;
            sV[c][u] = vb[(size_t)c * 1024 + u];
        }
#endif
        __syncthreads();

        // ---- scores: S[w][u] = sum_c Q[c][w]*K[c][u]  (A = Q^T, B = K) ----
        if (wv < 4) {
            const int smB = (wv & 1) * 16;
            const int snB = (wv >> 1) * 16;
            v8f s = {0.f,0.f,0.f,0.f,0.f,0.f,0.f,0.f};
            #pragma unroll
            for (int kk = 0; kk < 32; kk += 4) {
                v2f a, bf;
                a.x  = sQ[kk + 2 * hh    ][smB + l16];
                a.y  = sQ[kk + 2 * hh + 1][smB + l16];
                bf.x = sK[kk + 2 * hh    ][snB + l16];
                bf.y = sK[kk + 2 * hh + 1][snB + l16];
                s = __builtin_amdgcn_wmma_f32_16x16x4_f32(
                        false, a, false, bf, (short)0, s, false, false);
            }
            #pragma unroll
            for (int r = 0; r < 8; ++r)
                sS[smB + r + 8 * hh][snB + l16] = s[r];
        }
        __syncthreads();

        // ---- softmax stats over w per key column dx (faithful axis=2) ----
        if (tid < 15) {
            const int dx = tid;
            float m = -3.4e38f;
            for (int ww = 0; ww < 32; ++ww) {
                int   u = ww + dx - 7;
                float v = (u >= 0 && u < 32) ? sS[ww][u] : 0.f;  // padded -> score 0
                m = fmaxf(m, v);
            }
            float s = 0.f;
            for (int ww = 0; ww < 32; ++ww) {
                int   u = ww + dx - 7;
                float v = (u >= 0 && u < 32) ? sS[ww][u] : 0.f;
                s += __expf(v - m);
            }
            cmax[dx]  = m;
            crinv[dx] = 1.f / s;
        }
        __syncthreads();

        // ---- banded weight matrix sW[w][u] ----
        for (int idx = tid; idx < 1024; idx += 256) {
            int w_ = idx >> 5, u = idx & 31;
            int dx = u - w_ + 7;
            float val = 0.f;
            if (dx >= 0 && dx < 15)
                val = __expf(sS[w_][u] - cmax[dx]) * crinv[dx];
            sW[w_][u] = val;
        }
        __syncthreads();

        // ---- D[w][c] += sW[w][u] * V[c][u]   (A = sW, B = V^T) ----
        #pragma unroll
        for (int kk = 0; kk < 32; kk += 4) {
            v2f a, b0, b1;
            a.x  = sW[mBase + l16][kk + 2 * hh    ];
            a.y  = sW[mBase + l16][kk + 2 * hh + 1];
            b0.x = sV[n0 + l16][kk + 2 * hh    ];
            b0.y = sV[n0 + l16][kk + 2 * hh + 1];
            b1.x = sV[n1 + l16][kk + 2 * hh    ];
            b1.y = sV[n1 + l16][kk + 2 * hh + 1];
            vacc0 = __builtin_amdgcn_wmma_f32_16x16x4_f32(
                        false, a, false, b0, (short)0, vacc0, false, false);
            vacc1 = __builtin_amdgcn_wmma_f32_16x16x4_f32(
                        false, a, false, b1, (short)0, vacc1, false, false);
        }
        __syncthreads();   // protect sK/sV/sS/sW before next chunk
    }

    // feat[b, 128+c, h, w] = D[w][c]
    float* fb = feat + ((size_t)b * 256 + 128) * 1024 + (size_t)h * 32;
    #pragma unroll
    for (int r = 0; r < 8; ++r) {
        int w_ = mBase + r + 8 * hh;
        fb[(size_t)(n0 + l16) * 1024 + w_] = vacc0[r];
        fb[(size_t)(n1 + l16) * 1024 + w_] = vacc1[r];
    }
}

// ---------------------------------------------------------------------------
// Host orchestration. inputs: x, mem, w_qk, w_qv, w_mk, w_mv, w_smooth,
//                             gamma, beta, mean, var   (all fp32)
// ---------------------------------------------------------------------------
extern "C" void kernel_launch(void* const* d_in, const int* in_sizes, int n_in,
                              void* d_out, int out_size, void* d_ws, size_t ws_size,
                              hipStream_t stream)
{
    (void)in_sizes; (void)n_in; (void)out_size; (void)ws_size;
    const float* x        = (const float*)d_in[0];
    const float* mem      = (const float*)d_in[1];
    const float* w_qk     = (const float*)d_in[2];
    const float* w_qv     = (const float*)d_in[3];
    const float* w_mk     = (const float*)d_in[4];
    const float* w_mv     = (const float*)d_in[5];
    const float* w_smooth = (const float*)d_in[6];
    const float* bn_gamma = (const float*)d_in[7];
    const float* bn_beta  = (const float*)d_in[8];
    const float* bn_mean  = (const float*)d_in[9];
    const float* bn_var   = (const float*)d_in[10];
    float* out = (float*)d_out;
    float* ws  = (float*)d_ws;

    // workspace layout (floats)
    float* feat = ws;               // (2,256,1024) = 524288
    float* qkb  = ws + 524288;      // (2,32,1024)  = 65536
    float* mkb  = ws + 589824;      // (2,3,32,1024)= 196608
    float* mvb  = ws + 786432;      // (2,3,128,1024)=786432

    dim3 blk(128);

    // query_k = w_qk @ x
    gemm_wmma<<<dim3(16, 2, 2), blk, 0, stream>>>(w_qk, x, qkb, 256, 256, 1024, 1024,
                                                  1, 262144, 0, 32768, 0);
    // query_v = w_qv @ x -> feat channels [0,128)
    gemm_wmma<<<dim3(16, 8, 2), blk, 0, stream>>>(w_qv, x, feat, 256, 256, 1024, 1024,
                                                  1, 262144, 0, 262144, 0);
    // mem_k[t=0] = w_mk @ x
    gemm_wmma<<<dim3(16, 2, 2), blk, 0, stream>>>(w_mk, x, mkb, 256, 256, 1024, 1024,
                                                  1, 262144, 0, 98304, 0);
    // mem_k[t=1,2] = w_mk @ mem   (z = b*2+tt)
    gemm_wmma<<<dim3(16, 2, 4), blk, 0, stream>>>(w_mk, mem, mkb + 32768, 256, 256, 1024, 1024,
                                                  2, 524288, 262144, 98304, 32768);
    // mem_v[t=0] = w_mv @ x
    gemm_wmma<<<dim3(16, 8, 2), blk, 0, stream>>>(w_mv, x, mvb, 256, 256, 1024, 1024,
                                                  1, 262144, 0, 393216, 0);
    // mem_v[t=1,2] = w_mv @ mem
    gemm_wmma<<<dim3(16, 8, 4), blk, 0, stream>>>(w_mv, mem, mvb + 131072, 256, 256, 1024, 1024,
                                                  2, 524288, 262144, 393216, 131072);

    // fused windowed attention -> feat channels [128,256)
    attn_kernel<<<dim3(32, 2), dim3(256), 0, stream>>>(qkb, mkb, mvb, feat);

    // out = ReLU(BN(w_smooth @ feat))
    gemm_wmma_bn_relu<<<dim3(16, 16, 2), blk, 0, stream>>>(w_smooth, feat, out,
                                                           256, 256, 1024, 1024,
                                                           262144, 262144,
                                                           bn_gamma, bn_beta, bn_mean, bn_var);
}